// GIN_55370718380132
// MI455X (gfx1250) — compile-verified
//
#include <hip/hip_runtime.h>

typedef __attribute__((ext_vector_type(2))) float v2f;
typedef __attribute__((ext_vector_type(8))) float v8f;

#define DFEAT 64

// ---------------- Phase 0: zero the padded aggregation buffer ----------------
__global__ void gin_zero_kernel(float4* __restrict__ p, long n4) {
    long i = (long)blockIdx.x * blockDim.x + threadIdx.x;
    if (i < n4) p[i] = make_float4(0.f, 0.f, 0.f, 0.f);
}

// ---------------- Phase 1: msg = node_feat[src] + edge_feat; atomic scatter-add to agg[dst] ----
// 16 threads per edge, each handling one float4 chunk of the 64-wide feature.
// edge_feat read is fully coalesced; node_feat gather and agg atomics stay L2-resident.
__global__ void gin_scatter_kernel(const float* __restrict__ nodef,
                                   const float* __restrict__ edgef,
                                   const int* __restrict__ eidx,
                                   float* __restrict__ agg, int E) {
    long tid = (long)blockIdx.x * blockDim.x + threadIdx.x;
    long total = (long)E * 16;
    if (tid >= total) return;
    int e = (int)(tid >> 4);
    int c = ((int)tid & 15) << 2;
    int src = eidx[e];
    int dst = eidx[E + e];
    const float4 nf = *(const float4*)(nodef + (size_t)src * DFEAT + c);
    const float4 ef = *(const float4*)(edgef + (size_t)e * DFEAT + c);
    float* dp = agg + (size_t)dst * DFEAT + c;
    atomicAdd(dp + 0, nf.x + ef.x);
    atomicAdd(dp + 1, nf.y + ef.y);
    atomicAdd(dp + 2, nf.z + ef.z);
    atomicAdd(dp + 3, nf.w + ef.w);
}

// ---------------- Phase 2: fused MLP  out = relu(agg@W1 + b1)@W2 + b2 -------
// One wave32 per 16-row node tile; 8 waves (256 threads) per block = 128 rows.
// Weights are staged transposed in LDS so every WMMA B-fragment is a contiguous
// ds_load_b64. The ReLU'd 16x16 h-tile is staged through per-wave LDS to convert
// C/D layout -> A-fragment layout for the second GEMM (same-wave DS is in-order).
__global__ __launch_bounds__(256) void gin_mlp_kernel(
        const float* __restrict__ agg,
        const float* __restrict__ W1, const float* __restrict__ b1,
        const float* __restrict__ W2, const float* __restrict__ b2,
        float* __restrict__ out, int N) {
    extern __shared__ float lds[];
    float* w1t = lds;            // W1^T: [128][64]  (32 KB)
    float* w2t = lds + 8192;     // W2^T: [64][128]  (32 KB)
    float* hst = lds + 16384;    // per-wave h staging: 8 * 16*16 floats (8 KB)

    const int tid = threadIdx.x;
    // cooperative transposed weight load (coalesced global reads)
    for (int i = tid; i < 8192; i += 256) {           // W1 is [64][128] row-major
        int k = i >> 7, n = i & 127;
        w1t[n * 64 + k] = W1[i];
    }
    for (int i = tid; i < 8192; i += 256) {           // W2 is [128][64] row-major
        int k = i >> 6, n = i & 63;
        w2t[n * 128 + k] = W2[i];
    }
    __syncthreads();

    const int wave = tid >> 5;
    const int lane = tid & 31;
    const int hi   = lane >> 4;   // 0: lanes 0-15, 1: lanes 16-31
    const int ln   = lane & 15;
    const int row0 = (blockIdx.x * 8 + wave) * 16;

    // Preload GEMM1 A-fragments: lane holds A(M=ln, K=4s+2*hi .. +1) for s=0..15
    const float* ap = agg + (size_t)(row0 + ln) * DFEAT + 2 * hi;
    v2f a1[16];
#pragma unroll
    for (int s = 0; s < 16; ++s)
        a1[s] = *(const v2f*)(ap + 4 * s);

    v8f c2[4];
#pragma unroll
    for (int o = 0; o < 4; ++o) c2[o] = (v8f){};

    float* hs = hst + wave * 256;

    for (int j = 0; j < 8; ++j) {           // 8 tiles of the 128-wide hidden dim
        // GEMM1: c1[16,16] = agg_tile[16,64] x W1[:, 16j..16j+15]
        v8f c1 = (v8f){};
#pragma unroll
        for (int s = 0; s < 16; ++s) {
            // B-frag: lane holds W1(k=4s+2*hi .. +1, n=16j+ln) -> contiguous in W1^T
            v2f b = *(const v2f*)(&w1t[(16 * j + ln) * 64 + 4 * s + 2 * hi]);
            c1 = __builtin_amdgcn_wmma_f32_16x16x4_f32(
                false, a1[s], false, b, (short)0, c1, false, false);
        }
        // bias + ReLU (bias depends only on column n = ln within this tile)
        float bias = b1[16 * j + ln];
#pragma unroll
        for (int r = 0; r < 8; ++r) {
            float v = c1[r] + bias;
            c1[r] = v > 0.f ? v : 0.f;
        }
        // stage h_j tile to LDS in row-major [m][n]
#pragma unroll
        for (int r = 0; r < 8; ++r)
            hs[(r + 8 * hi) * 16 + ln] = c1[r];

        // GEMM2 partial: c2[16,64] += h_j[16,16] x W2[16j..16j+15, :]
#pragma unroll
        for (int t = 0; t < 4; ++t) {
            // A-frag from staged h: lane holds h(m=ln, k=4t+2*hi .. +1)
            v2f a2 = *(const v2f*)(&hs[ln * 16 + 4 * t + 2 * hi]);
#pragma unroll
            for (int o = 0; o < 4; ++o) {
                v2f bf = *(const v2f*)(&w2t[(16 * o + ln) * 128 + 16 * j + 4 * t + 2 * hi]);
                c2[o] = __builtin_amdgcn_wmma_f32_16x16x4_f32(
                    false, a2, false, bf, (short)0, c2[o], false, false);
            }
        }
    }

    // bias + store (guard tail rows; padded agg rows are zero so WMMA math was safe)
#pragma unroll
    for (int o = 0; o < 4; ++o) {
        float bias = b2[16 * o + ln];
#pragma unroll
        for (int r = 0; r < 8; ++r) {
            int row = row0 + r + 8 * hi;
            if (row < N)
                out[(size_t)row * DFEAT + 16 * o + ln] = c2[o][r] + bias;
        }
    }
}

extern "C" void kernel_launch(void* const* d_in, const int* in_sizes, int n_in,
                              void* d_out, int out_size, void* d_ws, size_t ws_size,
                              hipStream_t stream) {
    const float* node_feat = (const float*)d_in[0];
    const float* edge_feat = (const float*)d_in[1];
    const int*   eidx      = (const int*)d_in[2];
    const float* W1 = (const float*)d_in[3];
    const float* b1 = (const float*)d_in[4];
    const float* W2 = (const float*)d_in[5];
    const float* b2 = (const float*)d_in[6];
    float* out = (float*)d_out;

    const int N = in_sizes[0] / DFEAT;
    const int E = in_sizes[1] / DFEAT;
    const int NPAD = ((N + 127) / 128) * 128;   // pad so every wave owns a full 16-row tile

    float* agg = (float*)d_ws;                  // [NPAD][64] f32 accumulator

    // zero padded accumulator
    long n4 = (long)NPAD * (DFEAT / 4);
    gin_zero_kernel<<<(unsigned)((n4 + 255) / 256), 256, 0, stream>>>((float4*)agg, n4);

    // message + scatter-add
    long nth = (long)E * 16;
    gin_scatter_kernel<<<(unsigned)((nth + 255) / 256), 256, 0, stream>>>(
        node_feat, edge_feat, eidx, agg, E);

    // fused 2-layer MLP with f32 WMMA
    size_t shmem = (size_t)(16384 + 8 * 256) * sizeof(float);  // 72 KB dynamic LDS
    gin_mlp_kernel<<<NPAD / 128, 256, shmem, stream>>>(agg, W1, b1, W2, b2, out, N);
}